// DigitCaps_49709951484053
// MI455X (gfx1250) — compile-verified
//
#include <hip/hip_runtime.h>
#include <hip/hip_bf16.h>

// CapsNet DigitCaps dynamic routing for MI455X (gfx1250, wave32).
//
// Kernel 1: u_hat = einsum("iodk,bik->biod", W, u) as 1152 independent
//           [256x8]@[8x160] GEMMs using V_WMMA_F32_16X16X4_F32 (exact f32).
//           One wave per (input capsule i, 16-row batch tile); N-tile == one
//           output capsule (d == lane), 2 WMMA K-steps per tile, 20 per wave.
// Kernel 2: 3 routing iterations, one workgroup per batch element, logits/
//           coefficients in dynamic LDS (~94 KB < 320 KB/WGP), block barriers
//           between phases.

#define B_SZ     256
#define IN_CAPS  1152
#define IN_DIM   8
#define OUT_CAPS 10
#define OUT_DIM  16
#define ND       (OUT_CAPS * OUT_DIM)       // 160
#define IO       (IN_CAPS * OUT_CAPS)       // 11520
#define ROUTE_THREADS 320

typedef float v2f __attribute__((ext_vector_type(2)));
typedef float v8f __attribute__((ext_vector_type(8)));

// ---------------------------------------------------------------------------
// Kernel 1: u_hat via f32 WMMA. grid = (4, IN_CAPS), block = 128 (4 waves).
// Wave w of block x handles batch rows [ (x*4+w)*16, +16 ).
// ---------------------------------------------------------------------------
__global__ __launch_bounds__(128)
void uhat_wmma_kernel(const float* __restrict__ u,
                      const float* __restrict__ W,
                      float* __restrict__ uhat) {
    const int i    = blockIdx.y;                       // input capsule
    const int wave = threadIdx.x >> 5;
    const int lane = threadIdx.x & 31;
    const int b0   = (blockIdx.x * 4 + wave) * 16;     // batch tile start
    const int m    = lane & 15;                        // row / N index
    const int hi   = lane >> 4;                        // K half select

    // A-matrix (16x4 f32, ISA layout): VGPR0 = K={0|2}, VGPR1 = K={1|3}.
    // a0 covers k=0..3, a1 covers k=4..7 (K-step of 4 per WMMA).
    const float* up = u + ((size_t)(b0 + m) * IN_CAPS + i) * IN_DIM;
    const v2f a0 = *reinterpret_cast<const v2f*>(up + 2 * hi);
    const v2f a1 = *reinterpret_cast<const v2f*>(up + 4 + 2 * hi);

    const float* wi = W + (size_t)i * ND * IN_DIM;     // W[i,:,:,:]

    #pragma unroll
    for (int o = 0; o < OUT_CAPS; ++o) {
        // B-matrix (4x16 f32): row K striped across lanes, N = d = m.
        const float* wp = wi + (o * OUT_DIM + m) * IN_DIM;
        if (o < OUT_CAPS - 1)
            __builtin_prefetch(wp + OUT_DIM * IN_DIM, 0, 0);  // next capsule's W
        const v2f bb0 = *reinterpret_cast<const v2f*>(wp + 2 * hi);
        const v2f bb1 = *reinterpret_cast<const v2f*>(wp + 4 + 2 * hi);

        v8f acc = {};
        acc = __builtin_amdgcn_wmma_f32_16x16x4_f32(false, a0, false, bb0,
                                                    (short)0, acc, false, false);
        acc = __builtin_amdgcn_wmma_f32_16x16x4_f32(false, a1, false, bb1,
                                                    (short)0, acc, false, false);

        // C/D layout: VGPR r -> rows {r | r+8} split on lane halves, N = lane%16.
        // u_hat flat index: ((b*IN_CAPS + i)*OUT_CAPS + o)*OUT_DIM + d
        float* outp = uhat +
            (((size_t)(b0 + hi * 8) * IN_CAPS + i) * OUT_CAPS + o) * OUT_DIM + m;
        const size_t row_stride = (size_t)IN_CAPS * ND;  // +1 batch row
        #pragma unroll
        for (int r = 0; r < 8; ++r)
            outp[(size_t)r * row_stride] = acc[r];
    }
}

// ---------------------------------------------------------------------------
// Kernel 2: dynamic routing. grid = B_SZ, block = 320, dynamic LDS.
// LDS: b_log[11520] | c[11520] | s_part[320] | v[160] | fac[16]
// ---------------------------------------------------------------------------
__global__ __launch_bounds__(ROUTE_THREADS)
void routing_kernel(const float* __restrict__ uhat, float* __restrict__ out) {
    extern __shared__ float lds[];
    float* b_log  = lds;                 // [IN_CAPS][OUT_CAPS]
    float* c      = b_log + IO;          // [IN_CAPS][OUT_CAPS]
    float* s_part = c + IO;              // [2][160]
    float* v_sh   = s_part + 2 * ND;     // [160]
    float* fac    = v_sh + ND;           // [10] (padded)

    const int    bb   = blockIdx.x;
    const int    tid  = threadIdx.x;
    const size_t base = (size_t)bb * IN_CAPS * ND;

    for (int idx = tid; idx < IO; idx += ROUTE_THREADS) b_log[idx] = 0.0f;
    __syncthreads();

    for (int it = 0; it < 3; ++it) {
        // ---- c = softmax(b, axis=o) --------------------------------------
        for (int i = tid; i < IN_CAPS; i += ROUTE_THREADS) {
            float mx = -1e30f;
            #pragma unroll
            for (int o = 0; o < OUT_CAPS; ++o)
                mx = fmaxf(mx, b_log[i * OUT_CAPS + o]);
            float sum = 0.0f;
            #pragma unroll
            for (int o = 0; o < OUT_CAPS; ++o) {
                float e = __expf(b_log[i * OUT_CAPS + o] - mx);
                c[i * OUT_CAPS + o] = e;
                sum += e;
            }
            const float inv = 1.0f / sum;
            #pragma unroll
            for (int o = 0; o < OUT_CAPS; ++o)
                c[i * OUT_CAPS + o] *= inv;
        }
        __syncthreads();

        // ---- s[o,d] = sum_i c[i,o] * u_hat[i,o,d] ------------------------
        {
            const int slot = tid % ND;          // 160 (o,d) slots
            const int half = tid / ND;          // 2 halves of i-range
            const int o = slot >> 4, d = slot & 15;
            const int i0 = half * (IN_CAPS / 2), i1 = i0 + (IN_CAPS / 2);
            float acc = 0.0f;
            for (int i = i0; i < i1; ++i)
                acc = fmaf(c[i * OUT_CAPS + o],
                           uhat[base + ((size_t)i * OUT_CAPS + o) * OUT_DIM + d],
                           acc);
            s_part[half * ND + slot] = acc;
        }
        __syncthreads();
        if (tid < ND) v_sh[tid] = s_part[tid] + s_part[ND + tid];
        __syncthreads();

        // ---- squash ------------------------------------------------------
        if (tid < OUT_CAPS) {
            float sq = 0.0f;
            #pragma unroll
            for (int d = 0; d < OUT_DIM; ++d) {
                const float x = v_sh[tid * OUT_DIM + d];
                sq = fmaf(x, x, sq);
            }
            fac[tid] = (sq / (1.0f + sq)) * __frsqrt_rn(sq + 1e-8f);
        }
        __syncthreads();
        if (tid < ND) v_sh[tid] *= fac[tid >> 4];
        __syncthreads();

        // ---- b[i,o] += <u_hat[i,o,:], v[o,:]> ----------------------------
        if (it < 2) {
            for (int idx = tid; idx < IO; idx += ROUTE_THREADS) {
                const int o = idx % OUT_CAPS;
                const float* up = uhat + base + (size_t)idx * OUT_DIM;
                float dot = 0.0f;
                #pragma unroll
                for (int d = 0; d < OUT_DIM; ++d)
                    dot = fmaf(up[d], v_sh[o * OUT_DIM + d], dot);
                b_log[idx] += dot;
            }
            __syncthreads();
        }
    }

    if (tid < ND) out[(size_t)bb * ND + tid] = v_sh[tid];
}

// ---------------------------------------------------------------------------
extern "C" void kernel_launch(void* const* d_in, const int* in_sizes, int n_in,
                              void* d_out, int out_size, void* d_ws, size_t ws_size,
                              hipStream_t stream) {
    const float* u = (const float*)d_in[0];   // [256,1152,8]
    const float* W = (const float*)d_in[1];   // [1152,10,16,8]
    float* out  = (float*)d_out;              // [256,10,16]
    float* uhat = (float*)d_ws;               // [256,1152,10,16] = 189 MB scratch

    // Kernel 1: 1152 x 16 batch tiles, 4 waves per block.
    uhat_wmma_kernel<<<dim3(4, IN_CAPS), 128, 0, stream>>>(u, W, uhat);

    // Kernel 2: one workgroup per batch element; dynamic LDS.
    const size_t lds_bytes = (size_t)(2 * IO + 2 * ND + ND + 16) * sizeof(float);
    routing_kernel<<<dim3(B_SZ), ROUTE_THREADS, lds_bytes, stream>>>(uhat, out);
}